// compress_jpeg_66855460929545
// MI455X (gfx1250) — compile-verified
//
#include <hip/hip_runtime.h>
#include <math.h>

typedef float v2f __attribute__((ext_vector_type(2)));
typedef float v8f __attribute__((ext_vector_type(8)));

// Quantization tables stored exactly as the numpy literals (pre-transpose).
// The reference uses table.T, so divisor(u,v) = base[v][u] = kTab[v*8+u].
__constant__ float kYTab[64] = {
  16,11,10,16,24,40,51,61,
  12,12,14,19,26,58,60,55,
  14,13,16,24,40,57,69,56,
  14,17,22,29,51,87,80,62,
  18,22,37,56,68,109,103,77,
  24,35,55,64,81,104,113,92,
  49,64,78,87,103,121,120,101,
  72,92,95,98,112,100,103,99};

__constant__ float kCTab[64] = {
  17,18,24,47,99,99,99,99,
  18,21,26,66,99,99,99,99,
  24,26,56,99,99,99,99,99,
  47,66,99,99,99,99,99,99,
  99,99,99,99,99,99,99,99,
  99,99,99,99,99,99,99,99,
  99,99,99,99,99,99,99,99,
  99,99,99,99,99,99,99,99};

// LDS layout (floats), one 16B-aligned pool:
#define STRIDE_F 36            // full-res row stride (mult of 4 -> 16B-aligned v2f/v4f)
#define STRIDE_P 18            // pooled row stride
#define OFF_Y    0
#define OFF_CB   (32 * STRIDE_F)           // 1152
#define OFF_CR   (2 * 32 * STRIDE_F)       // 2304
#define OFF_CBP  (3 * 32 * STRIDE_F)       // 3456
#define OFF_CRP  (3 * 32 * STRIDE_F + 16 * STRIDE_P)  // 3744
#define SMEM_F   (3 * 32 * STRIDE_F + 2 * 16 * STRIDE_P) // 4032 floats = 16128 B

// One 16x16 tile = 2x2 grid of 8x8 blocks.
// D = blockdiag(C^T) * (X-128) * blockdiag(C), via two WMMA_F32_16X16X4 passes.
// Stage 1: Q = (X-128)*R  (A = X from LDS, B = R const; K-slot(v,h) -> y = 4kk+v+2h)
// Stage 2: D = L*Q        (A = L const, B = Q's D-layout VGPRs directly via the
//                          K-permutation slot(kk,v,h) -> m = 2kk+v+8h)
__device__ __forceinline__ void dct_quant_store(
    const float* __restrict__ xlds, int ldstride,
    const float (&Rb)[4][2], const float (&La)[4][2],
    const float (&coef)[8],                       // 0.25*au*av / (tab*factor)
    float* __restrict__ qout, float* __restrict__ dout,
    int bz, int NB, int BPR, int brow0, int bcol0, int lane)
{
  const int h = lane >> 4;
  const int m = lane & 15;

  v8f q = {0.f,0.f,0.f,0.f,0.f,0.f,0.f,0.f};
#pragma unroll
  for (int kk = 0; kk < 4; ++kk) {
    v2f a = *(const v2f*)(xlds + m * ldstride + 4 * kk + 2 * h);
    a = a - 128.0f;
    v2f bb; bb.x = Rb[kk][0]; bb.y = Rb[kk][1];
    q = __builtin_amdgcn_wmma_f32_16x16x4_f32(false, a, false, bb, (short)0, q,
                                              false, false);
  }

  v8f d = {0.f,0.f,0.f,0.f,0.f,0.f,0.f,0.f};
#pragma unroll
  for (int kk = 0; kk < 4; ++kk) {
    v2f aL; aL.x = La[kk][0]; aL.y = La[kk][1];
    v2f bq; bq.x = q[2 * kk]; bq.y = q[2 * kk + 1];   // Q VGPRs ARE the B operand
    d = __builtin_amdgcn_wmma_f32_16x16x4_f32(false, aL, false, bq, (short)0, d,
                                              false, false);
  }

  // D layout: VGPR r -> (u'=r, j=lane) lanes 0-15, (u'=r+8, j=lane-16) lanes 16-31.
  const int v  = lane & 7;
  const int qr = lane >> 4;
  const int qc = (lane >> 3) & 1;
  const int n  = (brow0 + qr) * BPR + (bcol0 + qc);
  const size_t obase = ((size_t)bz * (size_t)NB + (size_t)n) * 64 + (size_t)v;
#pragma unroll
  for (int r = 0; r < 8; ++r) {
    const float before = d[r] * coef[r];
    const float after  = rintf(before);           // RNE == jnp.round
    qout[obase + r * 8] = after;
    dout[obase + r * 8] = fabsf(after - before);
  }
}

__global__ __launch_bounds__(32) void jpeg_compress_kernel(
    const float* __restrict__ img, const float* __restrict__ factorp,
    float* __restrict__ out)
{
  __shared__ __align__(16) float smem[SMEM_F];
  float* sY   = smem + OFF_Y;
  float* sCB  = smem + OFF_CB;
  float* sCR  = smem + OFF_CR;
  float* sCBp = smem + OFF_CBP;
  float* sCRp = smem + OFF_CRP;

  const int lane = threadIdx.x;
  const int tx = blockIdx.x, ty = blockIdx.y, bz = blockIdx.z;
  const int R0 = ty * 32, C0 = tx * 32;
  const float factor = factorp[0];

  // ---- Per-lane constant WMMA operands (block-diagonal cosine matrices) ----
  const int n15 = lane & 15, hh = lane >> 4;
  const float PI16 = 0.19634954084936207f;  // pi/16
  float Rb[4][2], La[4][2];
#pragma unroll
  for (int kk = 0; kk < 4; ++kk) {
#pragma unroll
    for (int vv = 0; vv < 2; ++vv) {
      const int y = 4 * kk + vv + 2 * hh;     // stage-1 K-slot -> column y of X
      Rb[kk][vv] = ((y >> 3) == (n15 >> 3))
                       ? cosf((float)(2 * (y & 7) + 1) * (float)(n15 & 7) * PI16)
                       : 0.0f;
      const int mm = 2 * kk + vv + 8 * hh;    // stage-2 K-slot -> row m of Q
      La[kk][vv] = ((n15 >> 3) == (mm >> 3))
                       ? cosf((float)(2 * (mm & 7) + 1) * (float)(n15 & 7) * PI16)
                       : 0.0f;
    }
  }

  // ---- Hoisted quantization coefficients: 16 divisions total per lane ----
  const int vq = lane & 7;
  const float av = (vq == 0) ? 0.70710678118654752f : 1.0f;
  float coefY[8], coefC[8];
#pragma unroll
  for (int r = 0; r < 8; ++r) {
    const float au = (r == 0) ? 0.70710678118654752f : 1.0f;
    const float sc = 0.25f * au * av;
    coefY[r] = sc / (kYTab[vq * 8 + r] * factor);
    coefC[r] = sc / (kCTab[vq * 8 + r] * factor);
  }

  // ---- Phase 1: RGB -> YCbCr into LDS, float2 (packed FMA) path ----
  // x255 folded into the affine coefficients.
  const size_t plane = 1024ull * 1024ull;
  const float* pr = img + (size_t)bz * 3 * plane;
  const float* pg = pr + plane;
  const float* pb = pg + plane;
  const int prow = lane >> 4;          // 0..1
  const int pcol = (lane & 15) * 2;    // 0..30, even
  for (int i = 0; i < 16; ++i) {
    const int row = i * 2 + prow;
    const size_t off = (size_t)(R0 + row) * 1024 + (size_t)(C0 + pcol);
    const v2f r = *(const v2f*)(pr + off);
    const v2f g = *(const v2f*)(pg + off);
    const v2f b = *(const v2f*)(pb + off);
    const v2f y  = 76.245f * r + 149.685f * g + 29.07f * b;
    const v2f cb = -43.02768f * r - 84.47232f * g + 127.5f * b + 128.0f;
    const v2f cr = 127.5f * r - 106.76544f * g - 20.73456f * b + 128.0f;
    const int sidx = row * STRIDE_F + pcol;
    *(v2f*)(sY  + sidx) = y;
    *(v2f*)(sCB + sidx) = cb;
    *(v2f*)(sCR + sidx) = cr;
  }
  __syncthreads();

  // ---- Phase 1.5: 2x2 average pool of chroma (affine commutes with mean) ----
#pragma unroll
  for (int e = 0; e < 8; ++e) {
    const int idx = e * 32 + lane;
    const int pm = idx >> 4, pc = idx & 15;
    const int base = (2 * pm) * STRIDE_F + 2 * pc;
    const v2f cb2 = *(const v2f*)(sCB + base) + *(const v2f*)(sCB + base + STRIDE_F);
    const v2f cr2 = *(const v2f*)(sCR + base) + *(const v2f*)(sCR + base + STRIDE_F);
    sCBp[pm * STRIDE_P + pc] = 0.25f * (cb2.x + cb2.y);
    sCRp[pm * STRIDE_P + pc] = 0.25f * (cr2.x + cr2.y);
  }
  __syncthreads();

  // ---- Output layout: (y_q, cb_q, cr_q, y_d, cb_d, cr_d) concatenated ----
  const size_t SY = 16ull * 16384ull * 64ull;
  const size_t SC = 16ull * 4096ull * 64ull;
  float* yq  = out;
  float* cbq = out + SY;
  float* crq = out + SY + SC;
  float* yd  = out + SY + 2 * SC;
  float* cbd = out + 2 * SY + 2 * SC;
  float* crd = out + 2 * SY + 3 * SC;

  // ---- Phase 2: 4 luma 16x16 tiles (each = four 8x8 DCTs via WMMA) ----
#pragma unroll
  for (int t = 0; t < 4; ++t) {
    const int tr = t >> 1, tc = t & 1;
    dct_quant_store(sY + (tr * 16) * STRIDE_F + tc * 16, STRIDE_F, Rb, La, coefY,
                    yq, yd, bz, 16384, 128, ty * 4 + tr * 2, tx * 4 + tc * 2,
                    lane);
  }

  // ---- Phase 3: chroma tiles (pooled 16x16 each) ----
  dct_quant_store(sCBp, STRIDE_P, Rb, La, coefC, cbq, cbd, bz, 4096, 64,
                  ty * 2, tx * 2, lane);
  dct_quant_store(sCRp, STRIDE_P, Rb, La, coefC, crq, crd, bz, 4096, 64,
                  ty * 2, tx * 2, lane);
}

extern "C" void kernel_launch(void* const* d_in, const int* in_sizes, int n_in,
                              void* d_out, int out_size, void* d_ws, size_t ws_size,
                              hipStream_t stream) {
  const float* img     = (const float*)d_in[0];
  const float* factorp = (const float*)d_in[1];
  float* out = (float*)d_out;
  (void)in_sizes; (void)n_in; (void)out_size; (void)d_ws; (void)ws_size;

  dim3 grid(32, 32, 16);   // (tile-x, tile-y, batch): 32x32-pixel region per wave
  dim3 block(32, 1, 1);    // one wave32 per workgroup
  hipLaunchKernelGGL(jpeg_compress_kernel, grid, block, 0, stream,
                     img, factorp, out);
}